// Encoder_5033701671193
// MI455X (gfx1250) — compile-verified
//
#include <hip/hip_runtime.h>
#include <math.h>

typedef __attribute__((ext_vector_type(2))) float v2f;
typedef __attribute__((ext_vector_type(8))) float v8f;

#define NEG_SLOPE 0.2f

// ---------------------------------------------------------------------------
// C[N,M] = A[N,K] @ W[K,M], fp32, via V_WMMA_F32_16X16X4_F32.
// One wave computes a full 16xM strip of C (M/16 accumulators), so each A
// element is loaded from global exactly once.  W is staged in LDS once per
// workgroup (max 128*128*4B = 64KB, fits in the 320KB WGP LDS) and all
// B-operand fetches are ds_loads.
// A layout (16x4, 32-bit): lanes 0-15 hold row M=lane, K={0,1}; lanes 16-31
// hold row M=lane-16, K={2,3}.  B (4x16) mirrored.  C/D: VGPR r, lanes 0-15
// -> M=r, N=lane; lanes 16-31 -> M=8+r, N=lane-16.
// ---------------------------------------------------------------------------
template <int K, int M>
__global__ void gemm_wmma_lds(const float* __restrict__ A,
                              const float* __restrict__ W,
                              float* __restrict__ C,
                              int Nrows) {
  constexpr int TILES_M = M / 16;
  __shared__ float wlds[K * M];

  // cooperative W -> LDS (all threads participate before any early exit)
  for (int i = threadIdx.x; i < K * M; i += blockDim.x) wlds[i] = W[i];
  __syncthreads();

  const int wavesPerBlock = blockDim.x >> 5;
  const int rowTile = blockIdx.x * wavesPerBlock + (threadIdx.x >> 5);
  const int lane = threadIdx.x & 31;
  if (rowTile >= (Nrows >> 4)) return;      // wave-uniform: EXEC all-1 past here

  const int l16 = lane & 15;
  const int kq  = (lane >> 4) << 1;         // 0 or 2
  const float* __restrict__ arow = A + (size_t)(rowTile * 16 + l16) * (size_t)K;

  v8f acc[TILES_M];
#pragma unroll
  for (int t = 0; t < TILES_M; ++t) acc[t] = (v8f){0.f,0.f,0.f,0.f,0.f,0.f,0.f,0.f};

  for (int k0 = 0; k0 < K; k0 += 4) {
    v2f a;
    a.x = arow[k0 + kq];
    a.y = arow[k0 + kq + 1];
    const float* wr0 = &wlds[(k0 + kq) * M + l16];
#pragma unroll
    for (int t = 0; t < TILES_M; ++t) {
      v2f b;
      b.x = wr0[t * 16];
      b.y = wr0[M + t * 16];
      acc[t] = __builtin_amdgcn_wmma_f32_16x16x4_f32(
          /*neg_a=*/false, a, /*neg_b=*/false, b,
          /*c_mod=*/(short)0, acc[t], /*reuse_a=*/false, /*reuse_b=*/false);
    }
  }

  const int rbase = rowTile * 16 + ((lane >> 4) << 3);
#pragma unroll
  for (int t = 0; t < TILES_M; ++t) {
    const int col = t * 16 + l16;
#pragma unroll
    for (int r = 0; r < 8; ++r)
      C[(size_t)(rbase + r) * (size_t)M + col] = acc[t][r];
  }
}

// ---------------------------------------------------------------------------
// als[n,h] = sum_c hs[n,h,c]*a_s[h,c]; ald likewise from hd/a_d.
// ---------------------------------------------------------------------------
__global__ void node_alpha(const float* __restrict__ hs,
                           const float* __restrict__ hd,
                           const float* __restrict__ a_s,
                           const float* __restrict__ a_d,
                           float* __restrict__ als,
                           float* __restrict__ ald, int Nn) {
  int tid = blockIdx.x * blockDim.x + threadIdx.x;
  if (tid >= Nn * 2) return;
  int n = tid >> 1, h = tid & 1;
  const float* hsrow = hs + (size_t)n * 128 + h * 64;
  const float* hdrow = hd + (size_t)n * 128 + h * 64;
  const float* asv = a_s + h * 64;
  const float* adv = a_d + h * 64;
  float s1 = 0.f, s2 = 0.f;
#pragma unroll 8
  for (int c = 0; c < 64; ++c) {
    s1 += hsrow[c] * asv[c];
    s2 += hdrow[c] * adv[c];
  }
  als[tid] = s1;
  ald[tid] = s2;
}

__global__ void init_layer(float* __restrict__ m, float* __restrict__ den,
                           float* __restrict__ agg, int Nn) {
  int tid = blockIdx.x * blockDim.x + threadIdx.x;
  if (tid < Nn * 64) agg[tid] = 0.f;
  if (tid < Nn * 2) { m[tid] = -INFINITY; den[tid] = 0.f; }
}

// float atomic max via sign-split int/uint atomics (native global_atomic_max_i32
// / global_atomic_min_u32; correct for mixed signs with -inf init).
__device__ inline void atomicMaxF(float* addr, float value) {
  if (value >= 0.f)
    atomicMax((int*)addr, __float_as_int(value));
  else
    atomicMin((unsigned int*)addr, __float_as_uint(value));
}

__global__ void edge_max(const int* __restrict__ src, const int* __restrict__ dst,
                         const float* __restrict__ als, const float* __restrict__ ald,
                         float* __restrict__ e, float* __restrict__ m, int Ee) {
  int tid = blockIdx.x * blockDim.x + threadIdx.x;
  if (tid >= Ee * 2) return;
  int ed = tid >> 1, h = tid & 1;
  int s = src[ed], d = dst[ed];
  float v = als[s * 2 + h] + ald[d * 2 + h];
  v = (v > 0.f) ? v : NEG_SLOPE * v;            // leaky_relu
  e[tid] = v;
  atomicMaxF(&m[d * 2 + h], v);
}

__global__ void edge_den(const int* __restrict__ dst,
                         const float* __restrict__ e, const float* __restrict__ m,
                         float* __restrict__ den, int Ee) {
  int tid = blockIdx.x * blockDim.x + threadIdx.x;
  if (tid >= Ee * 2) return;
  int ed = tid >> 1, h = tid & 1;
  int d = dst[ed];
  float ex = __expf(e[tid] - m[d * 2 + h]);
  atomicAdd(&den[d * 2 + h], ex);
}

// One wave per edge: agg[dst] += 0.5*(alpha0*hs[src,0,:] + alpha1*hs[src,1,:])
// (head-mean folded in -> only 64 atomic adds per edge).
__global__ void edge_scatter(const int* __restrict__ src, const int* __restrict__ dst,
                             const float* __restrict__ e, const float* __restrict__ m,
                             const float* __restrict__ den,
                             const float* __restrict__ hs,
                             float* __restrict__ agg, int Ee) {
  int wave = (blockIdx.x * blockDim.x + threadIdx.x) >> 5;
  int lane = threadIdx.x & 31;
  if (wave >= Ee) return;
  int s = src[wave], d = dst[wave];
  float e0 = e[wave * 2 + 0], e1 = e[wave * 2 + 1];
  float m0 = m[d * 2 + 0], m1 = m[d * 2 + 1];
  float n0 = den[d * 2 + 0], n1 = den[d * 2 + 1];
  float a0 = __expf(e0 - m0) / (n0 + 1e-16f) * 0.5f;
  float a1 = __expf(e1 - m1) / (n1 + 1e-16f) * 0.5f;
  const float* hsrow = hs + (size_t)s * 128;
  float v0 = hsrow[lane]      * a0 + hsrow[64 + lane] * a1;
  float v1 = hsrow[32 + lane] * a0 + hsrow[96 + lane] * a1;
  float* aggrow = agg + (size_t)d * 64;
  atomicAdd(&aggrow[lane],      v0);
  atomicAdd(&aggrow[32 + lane], v1);
}

__global__ void finalize(const float* __restrict__ agg, const float* __restrict__ hl,
                         const float* __restrict__ b, const float* __restrict__ bl,
                         float* __restrict__ out, int Nn) {
  int tid = blockIdx.x * blockDim.x + threadIdx.x;
  if (tid >= Nn * 64) return;
  int c = tid & 63;
  float v = agg[tid] + b[c] + hl[tid] + bl[c];
  out[tid] = (v > 0.f) ? v : 0.f;               // relu
}

// ---------------------------------------------------------------------------
extern "C" void kernel_launch(void* const* d_in, const int* in_sizes, int n_in,
                              void* d_out, int out_size, void* d_ws, size_t ws_size,
                              hipStream_t stream) {
  const float* x  = (const float*)d_in[0];
  const int*   ei = (const int*)d_in[1];
  const int Nn = in_sizes[0] / 128;   // 50000
  const int Ee = in_sizes[1] / 2;     // 800000
  const int* src = ei;
  const int* dst = ei + Ee;

  // per-layer params: Ws, Wd, as, ad, b, Wl, bl at indices 2..8, 9..15, 16..22
  const float *Ws[3], *Wd[3], *As[3], *Ad[3], *B[3], *Wl[3], *Bl[3];
  for (int l = 0; l < 3; ++l) {
    int base = 2 + 7 * l;
    Ws[l] = (const float*)d_in[base + 0];
    Wd[l] = (const float*)d_in[base + 1];
    As[l] = (const float*)d_in[base + 2];
    Ad[l] = (const float*)d_in[base + 3];
    B[l]  = (const float*)d_in[base + 4];
    Wl[l] = (const float*)d_in[base + 5];
    Bl[l] = (const float*)d_in[base + 6];
  }

  // workspace carve-out (fp32)
  float* ws = (float*)d_ws;
  size_t off = 0;
  float* hs  = ws + off; off += (size_t)Nn * 128;
  float* hd  = ws + off; off += (size_t)Nn * 128;
  float* hl  = ws + off; off += (size_t)Nn * 64;
  float* als = ws + off; off += (size_t)Nn * 2;
  float* ald = ws + off; off += (size_t)Nn * 2;
  float* m   = ws + off; off += (size_t)Nn * 2;
  float* den = ws + off; off += (size_t)Nn * 2;
  float* ebuf= ws + off; off += (size_t)Ee * 2;
  float* agg = ws + off; off += (size_t)Nn * 64;
  float* h1  = ws + off; off += (size_t)Nn * 64;
  float* h2  = ws + off; off += (size_t)Nn * 64;

  const float* lin[3]  = { x, h1, h2 };
  float*       lout[3] = { h1, h2, (float*)d_out };

  const int TPB = 256;
  const int rowTiles = Nn >> 4;                       // 3125
  const int gemmBlocks = (rowTiles + 7) / 8;          // 8 waves / block

  for (int l = 0; l < 3; ++l) {
    // GEMMs: hs = in@Ws [N,128], hd = in@Wd [N,128], hl = in@Wl [N,64]
    if (l == 0) {
      gemm_wmma_lds<128, 128><<<gemmBlocks, TPB, 0, stream>>>(lin[l], Ws[l], hs, Nn);
      gemm_wmma_lds<128, 128><<<gemmBlocks, TPB, 0, stream>>>(lin[l], Wd[l], hd, Nn);
      gemm_wmma_lds<128, 64 ><<<gemmBlocks, TPB, 0, stream>>>(lin[l], Wl[l], hl, Nn);
    } else {
      gemm_wmma_lds<64, 128><<<gemmBlocks, TPB, 0, stream>>>(lin[l], Ws[l], hs, Nn);
      gemm_wmma_lds<64, 128><<<gemmBlocks, TPB, 0, stream>>>(lin[l], Wd[l], hd, Nn);
      gemm_wmma_lds<64, 64 ><<<gemmBlocks, TPB, 0, stream>>>(lin[l], Wl[l], hl, Nn);
    }
    node_alpha<<<(Nn * 2 + TPB - 1) / TPB, TPB, 0, stream>>>(hs, hd, As[l], Ad[l], als, ald, Nn);
    init_layer<<<(Nn * 64 + TPB - 1) / TPB, TPB, 0, stream>>>(m, den, agg, Nn);
    edge_max<<<(Ee * 2 + TPB - 1) / TPB, TPB, 0, stream>>>(src, dst, als, ald, ebuf, m, Ee);
    edge_den<<<(Ee * 2 + TPB - 1) / TPB, TPB, 0, stream>>>(dst, ebuf, m, den, Ee);
    {
      long long threads = (long long)Ee * 32;
      edge_scatter<<<(int)((threads + TPB - 1) / TPB), TPB, 0, stream>>>(
          src, dst, ebuf, m, den, hs, agg, Ee);
    }
    finalize<<<(Nn * 64 + TPB - 1) / TPB, TPB, 0, stream>>>(agg, hl, B[l], Bl[l], lout[l], Nn);
  }
}